// Transformer_F_16630113370794
// MI455X (gfx1250) — compile-verified
//
#include <hip/hip_runtime.h>

typedef __attribute__((ext_vector_type(2))) float v2f;
typedef __attribute__((ext_vector_type(8))) float v8f;

#define NP1   1024
#define NTOK  1023
#define DP1   128
#define DRAW  127
#define NHEAD 8
#define TPAD  132   // padded staging row stride (floats): 528 B/row, 16B-aligned
#define QPAD  130   // padded P/Q row stride (floats): bank step 2 for strided reads

__device__ __forceinline__ v8f wmma_f32(v2f a, v2f b, v8f c) {
    return __builtin_amdgcn_wmma_f32_16x16x4_f32(
        /*neg_a=*/false, a, /*neg_b=*/false, b,
        /*c_mod=*/(short)0, c, /*reuse_a=*/false, /*reuse_b=*/false);
}

// P padded from [127,127] to [128,128] with P[127][127] = 1
__device__ __forceinline__ float padP(const float* __restrict__ p, int r, int c) {
    if (r < DRAW && c < DRAW) return p[r * DRAW + c];
    return (r == DRAW && c == DRAW) ? 1.0f : 0.0f;
}
// Q padded with zeros
__device__ __forceinline__ float padQ(const float* __restrict__ q, int r, int c) {
    return (r < DRAW && c < DRAW) ? q[r * DRAW + c] : 0.0f;
}

// Async-copy one 32x128 f32 tile (rows [grow_base, grow_base+32)) of Z into an
// LDS buffer: 256 lanes x 16B x 4 rounds, ASYNCcnt += 4 per wave.
__device__ __forceinline__ void stage_async(const float* __restrict__ gbase,
                                            float* ldsbuf, int tid, int grow_base) {
#pragma unroll
    for (int r = 0; r < 4; ++r) {
        const int idx = tid + r * 256;            // float4 index 0..1023
        const int row = idx >> 5;                 // 0..31
        const int c4  = (idx & 31) << 2;          // 0,4,...,124
        const unsigned lds_byte = (unsigned)(uintptr_t)(ldsbuf + row * TPAD + c4);
        const unsigned goff     = (unsigned)(((grow_base + row) * DP1 + c4) * 4);
        asm volatile("global_load_async_to_lds_b128 %0, %1, %2"
                     :: "v"(lds_byte), "v"(goff), "s"(gbase)
                     : "memory");
    }
}

// One transformer layer: Zout = Zin + Zin * (sum_h Q_h G P_h^T) / N,
// G = Zin[0:N]^T Zin[0:N].   grid = B blocks, 256 threads (8 waves).
__global__ __launch_bounds__(256)
void layer_kernel(const float* __restrict__ Zin,
                  const float* __restrict__ prm,   // [H][2][127][127] for this layer
                  float* __restrict__ Zout) {
    __shared__ __align__(16) float lds_tileA[32 * TPAD];  // double-buffered staging
    __shared__ __align__(16) float lds_tileB[32 * TPAD];
    __shared__ float lds_G[DP1 * DP1];        // Gram matrix
    __shared__ float lds_M[DP1 * DP1];        // per-head M, then final T/N
    __shared__ float lds_PT[DP1 * QPAD];      // padded P^T for current head
    __shared__ float lds_Q[DP1 * QPAD];       // padded Q  for current head

    const int b    = blockIdx.x;
    const int tid  = threadIdx.x;
    const int w    = tid >> 5;
    const int lane = tid & 31;
    const int half = lane >> 4;
    const int l16  = lane & 15;
    const int i0   = w * 16;                  // wave's 16-row strip base

    const float* Zb = Zin + (size_t)b * NP1 * DP1;
    float*      Zob = Zout + (size_t)b * NP1 * DP1;

    // Warm L2/L0 with this layer's parameters while phase 1 runs.
    {
        const char* pc = (const char*)prm;
        const int pbytes = NHEAD * 2 * DRAW * DRAW * 4;
        for (int off = tid * 64; off < pbytes; off += 256 * 64)
            __builtin_prefetch(pc + off, 0, 1);
    }

    // ---------- Phase 1: G = Z[0:1023]^T Z[0:1023] ----------
    v8f accG[8] = {};
    stage_async(Zb, lds_tileA, tid, 0);
    for (int tile = 0; tile < 32; ++tile) {
        float* cur = (tile & 1) ? lds_tileB : lds_tileA;
        float* nxt = (tile & 1) ? lds_tileA : lds_tileB;
        if (tile < 31) {
            stage_async(Zb, nxt, tid, (tile + 1) * 32);   // prefetch next tile
            asm volatile("s_wait_asynccnt 0x4" ::: "memory");
        } else {
            asm volatile("s_wait_asynccnt 0x0" ::: "memory");
        }
        __syncthreads();
        if (tile == 31) {               // mask token N: zero LDS row for global row 1023
            if (tid < 32) {
                float* dst = cur + 31 * TPAD + tid * 4;
                dst[0] = 0.f; dst[1] = 0.f; dst[2] = 0.f; dst[3] = 0.f;
            }
            __syncthreads();
        }
        for (int k = 0; k < 32; k += 4) {
            const int kr = k + (half << 1);   // this half-wave's K rows
            v2f a;
            a.x = cur[kr * TPAD + i0 + l16];
            a.y = cur[(kr + 1) * TPAD + i0 + l16];
#pragma unroll
            for (int t = 0; t < 8; ++t) {
                v2f bb;
                bb.x = cur[kr * TPAD + t * 16 + l16];
                bb.y = cur[(kr + 1) * TPAD + t * 16 + l16];
                accG[t] = wmma_f32(a, bb, accG[t]);
            }
        }
        __syncthreads();
    }
#pragma unroll
    for (int t = 0; t < 8; ++t)
#pragma unroll
        for (int r = 0; r < 8; ++r)
            lds_G[(i0 + r + 8 * half) * DP1 + t * 16 + l16] = accG[t][r];

    // ---------- Phase 2: T = sum_h Q_h (G P_h^T) ----------
    v8f accT[8] = {};
    for (int h = 0; h < NHEAD; ++h) {
        const float* Ph = prm + (size_t)h * 2 * DRAW * DRAW;
        const float* Qh = Ph + DRAW * DRAW;

        // Stage padded Q (row-major) and P^T into LDS; source reads coalesced,
        // pad cells (incl. P[127][127]=1 corner) materialized here.
        __syncthreads();            // prior readers of lds_PT/lds_Q/lds_M done
        for (int idx = tid; idx < DP1 * DP1; idx += 256) {
            const int r = idx >> 7;
            const int c = idx & (DP1 - 1);
            lds_Q[r * QPAD + c]  = padQ(Qh, r, c);
            lds_PT[c * QPAD + r] = padP(Ph, r, c);   // PT[c][r] = P[r][c]
        }
        __syncthreads();

        // M = G * P_h^T  (use G symmetry: A[j,c] = G[c,j] -> stride-1 lane reads)
        v8f accM[8] = {};
        for (int k = 0; k < DP1; k += 4) {
            const int kr = k + (half << 1);
            v2f a;
            a.x = lds_G[kr * DP1 + i0 + l16];
            a.y = lds_G[(kr + 1) * DP1 + i0 + l16];
#pragma unroll
            for (int t = 0; t < 8; ++t) {
                const int ip = t * 16 + l16;
                v2f bb;
                bb.x = lds_PT[kr * QPAD + ip];
                bb.y = lds_PT[(kr + 1) * QPAD + ip];
                accM[t] = wmma_f32(a, bb, accM[t]);
            }
        }
#pragma unroll
        for (int t = 0; t < 8; ++t)
#pragma unroll
            for (int r = 0; r < 8; ++r)
                lds_M[(i0 + r + 8 * half) * DP1 + t * 16 + l16] = accM[t][r];
        __syncthreads();

        // accT += Q_h * M
        for (int k = 0; k < DP1; k += 4) {
            const int kr = k + (half << 1);
            const int qrow = (i0 + l16) * QPAD;
            v2f a;
            a.x = lds_Q[qrow + kr];
            a.y = lds_Q[qrow + kr + 1];
#pragma unroll
            for (int t = 0; t < 8; ++t) {
                v2f bb;
                bb.x = lds_M[kr * DP1 + t * 16 + l16];
                bb.y = lds_M[(kr + 1) * DP1 + t * 16 + l16];
                accT[t] = wmma_f32(a, bb, accT[t]);
            }
        }
    }
    __syncthreads();
    const float invN = 1.0f / (float)NTOK;
#pragma unroll
    for (int t = 0; t < 8; ++t)
#pragma unroll
        for (int r = 0; r < 8; ++r)
            lds_M[(i0 + r + 8 * half) * DP1 + t * 16 + l16] = accT[t][r] * invN;
    __syncthreads();

    // ---------- Phase 3: Zout = Zin + Zin * (T/N) ----------
    const int rw = w >> 2;                     // 0/1: 16-row subtile of the 32-row tile
    const int cw = w & 3;                      // 0..3: pair of 16-col tiles
    stage_async(Zb, lds_tileA, tid, 0);
    for (int tile = 0; tile < 32; ++tile) {
        float* cur = (tile & 1) ? lds_tileB : lds_tileA;
        float* nxt = (tile & 1) ? lds_tileA : lds_tileB;
        const int base = tile * 32;
        if (tile < 31) {
            stage_async(Zb, nxt, tid, base + 32);
            asm volatile("s_wait_asynccnt 0x4" ::: "memory");
        } else {
            asm volatile("s_wait_asynccnt 0x0" ::: "memory");
        }
        __syncthreads();
        v8f acc[2] = {};
        for (int k = 0; k < DP1; k += 4) {
            const int kr = k + (half << 1);
            const int lrow = rw * 16 + l16;
            v2f a;
            a.x = cur[lrow * TPAD + kr];       // padded stride => no bank conflict
            a.y = cur[lrow * TPAD + kr + 1];
#pragma unroll
            for (int t = 0; t < 2; ++t) {
                const int cc = (cw * 2 + t) * 16 + l16;
                v2f bb;
                bb.x = lds_M[kr * DP1 + cc];
                bb.y = lds_M[(kr + 1) * DP1 + cc];
                acc[t] = wmma_f32(a, bb, acc[t]);
            }
        }
#pragma unroll
        for (int t = 0; t < 2; ++t) {
            const int cc = (cw * 2 + t) * 16 + l16;
#pragma unroll
            for (int r = 0; r < 8; ++r) {
                const int lrow = rw * 16 + r + 8 * half;
                Zob[(size_t)(base + lrow) * DP1 + cc] =
                    cur[lrow * TPAD + cc] + acc[t][r];
            }
        }
        __syncthreads();
    }
}

extern "C" void kernel_launch(void* const* d_in, const int* in_sizes, int n_in,
                              void* d_out, int out_size, void* d_ws, size_t ws_size,
                              hipStream_t stream) {
    const float* Z        = (const float*)d_in[0];   // [8,1024,128] f32
    const float* allparam = (const float*)d_in[1];   // [4,8,2,127,127] f32
    float* out = (float*)d_out;                      // [8,1024,128] f32
    float* ws  = (float*)d_ws;                       // >= 4 MB scratch (layer ping-pong)

    const size_t PARAM_L = (size_t)NHEAD * 2 * DRAW * DRAW;
    dim3 grid(8), block(256);
    // 4 sequentially-dependent layers; ping-pong ws <-> out
    layer_kernel<<<grid, block, 0, stream>>>(Z,   allparam + 0 * PARAM_L, ws);
    layer_kernel<<<grid, block, 0, stream>>>(ws,  allparam + 1 * PARAM_L, out);
    layer_kernel<<<grid, block, 0, stream>>>(out, allparam + 2 * PARAM_L, ws);
    layer_kernel<<<grid, block, 0, stream>>>(ws,  allparam + 3 * PARAM_L, out);
}